// ParticleModel_75806172774986
// MI455X (gfx1250) — compile-verified
//
#include <hip/hip_runtime.h>

#define NMAX 4   // DIM_M
#define NQ   5   // DIM_Q

typedef float v2f __attribute__((ext_vector_type(2)));
typedef float v4f __attribute__((ext_vector_type(4)));
typedef float v8f __attribute__((ext_vector_type(8)));

// ---------------------------------------------------------------------------
// Kernel A: fold weights.  H = W_cob^T * (W_reg * W_cob), all 4x4, computed
// with two V_WMMA_F32_16X16X4_F32 ops (4x4 blocks embedded in 16x16x4).
// Single wave32, EXEC all-ones at each WMMA (ISA 7.12 requirement).
//
// f32 16x4 A layout (ISA 7.12.2): lane L holds A[L%16, k] with k = 2*(L/16)
// in VGPR0 and k+1 in VGPR1.  B (4x16) mirrors: lanes 0-15 rows k=0,1;
// lanes 16-31 rows k=2,3; column n = L%16.
// D (16x16 f32): D[M,N] in VGPR M%8, lane N + 16*(M/8).
// ---------------------------------------------------------------------------
__global__ __launch_bounds__(32) void fold_weights_wmma(
    const float* __restrict__ W_cob,   // [4][4] row-major (C,F)
    const float* __restrict__ W_reg,   // [4][4] row-major (M,C)
    float* __restrict__ H)             // out: [4][4] row-major
{
  __shared__ float tmat[16];           // T = W_reg @ W_cob, 4x4 row-major
  const int L  = threadIdx.x;          // 0..31
  const int m  = L & 15;               // row (A) / column (B) index
  const int k0 = (L >> 4) * 2;         // K pair handled by this lane half
  const int m4 = m & 3;
  const float rowMask = (m < 4) ? 1.0f : 0.0f;

  // --- WMMA #1: T = W_reg (A, 16x4) x W_cob (B, 4x16) ---------------------
  v2f a1;                               // A[M,K] = W_reg[M,K] (M<4 else 0)
  a1.x = rowMask * W_reg[m4 * 4 + k0];
  a1.y = rowMask * W_reg[m4 * 4 + k0 + 1];
  v2f b1;                               // B[K,N] = W_cob[K,N] (N<4 else 0)
  b1.x = rowMask * W_cob[k0 * 4 + m4];
  b1.y = rowMask * W_cob[(k0 + 1) * 4 + m4];
  v8f cz = {};
  v8f T = __builtin_amdgcn_wmma_f32_16x16x4_f32(
      false, a1, false, b1, (short)0, cz, false, false);

  // T rows 0..3 live in D elements 0..3, lanes 0..3 (lane = column).
  if (L < 4) {
    tmat[0 * 4 + L] = T[0];
    tmat[1 * 4 + L] = T[1];
    tmat[2 * 4 + L] = T[2];
    tmat[3 * 4 + L] = T[3];
  }
  __syncthreads();

  // --- WMMA #2: H = W_cob^T (A2, 16x4) x T (B2, 4x16) ---------------------
  v2f a2;                               // A2[f,i] = W_cob[i,f]
  a2.x = rowMask * W_cob[k0 * 4 + m4];
  a2.y = rowMask * W_cob[(k0 + 1) * 4 + m4];
  v2f b2;                               // B2[i,f'] = T[i,f']
  b2.x = rowMask * tmat[k0 * 4 + m4];
  b2.y = rowMask * tmat[(k0 + 1) * 4 + m4];
  v8f Hd = __builtin_amdgcn_wmma_f32_16x16x4_f32(
      false, a2, false, b2, (short)0, cz, false, false);

  if (L < 4) {
    H[0 * 4 + L] = Hd[0];
    H[1 * 4 + L] = Hd[1];
    H[2 * 4 + L] = Hd[2];
    H[3 * 4 + L] = Hd[3];
  }
}

// ---------------------------------------------------------------------------
// Kernel B: per-block (256 batches) sum of valid counts.
// ---------------------------------------------------------------------------
__global__ __launch_bounds__(256) void block_count(
    const int* __restrict__ seq, int* __restrict__ sums, int Btot)
{
  __shared__ int red[256];
  const int tid = threadIdx.x;
  const int b = blockIdx.x * 256 + tid;
  int len = 0;
  if (b < Btot) {
    len = seq[b];
    len = len < 0 ? 0 : (len > NMAX ? NMAX : len);
  }
  red[tid] = len;
  __syncthreads();
  for (int off = 128; off > 0; off >>= 1) {
    if (tid < off) red[tid] += red[tid + off];
    __syncthreads();
  }
  if (tid == 0) sums[blockIdx.x] = red[0];
}

// ---------------------------------------------------------------------------
// Kernel C: single-block exclusive scan of block sums (in place), with carry
// across 1024-element chunks.  n ~ 7813 -> 8 chunks.
// ---------------------------------------------------------------------------
__global__ __launch_bounds__(1024) void scan_blocks(int* __restrict__ sums, int n)
{
  __shared__ int sm[1024];
  const int tid = threadIdx.x;
  int carry = 0;
  for (int base = 0; base < n; base += 1024) {
    const int i = base + tid;
    const int v = (i < n) ? sums[i] : 0;
    sm[tid] = v;
    __syncthreads();
    for (int off = 1; off < 1024; off <<= 1) {
      const int t = (tid >= off) ? sm[tid - off] : 0;
      __syncthreads();
      sm[tid] += t;
      __syncthreads();
    }
    const int incl  = sm[tid];
    const int total = sm[1023];
    __syncthreads();
    if (i < n) sums[i] = incl - v + carry;
    carry += total;
  }
}

// ---------------------------------------------------------------------------
// Kernel D: streaming kernel.  One thread per batch: 80B NT load, ~60 FLOPs,
// in-block LDS scan for compaction offsets, len NT stores.
// ---------------------------------------------------------------------------
__global__ __launch_bounds__(256) void particle_main(
    const float* __restrict__ xss,
    const int* __restrict__ seq,
    const float* __restrict__ H,         // 16 floats (uniform)
    const int* __restrict__ blockOff,    // exclusive block offsets
    float* __restrict__ out, int Btot)
{
  __shared__ int sc[256];
  const int tid = threadIdx.x;
  const int b = blockIdx.x * 256 + tid;

  int len = 0;
  if (b < Btot) {
    len = seq[b];
    len = len < 0 ? 0 : (len > NMAX ? NMAX : len);
  }

  // In-block exclusive scan (Hillis-Steele over 256).
  sc[tid] = len;
  __syncthreads();
  for (int off = 1; off < 256; off <<= 1) {
    const int t = (tid >= off) ? sc[tid - off] : 0;
    __syncthreads();
    sc[tid] += t;
    __syncthreads();
  }
  const int excl = sc[tid] - len;

  if (b >= Btot || len == 0) return;   // no further barriers below

  // 20 floats = 5 x b128, non-temporal (read-once stream).
  const v4f* p = (const v4f*)(xss + (size_t)b * (NMAX * NQ));
  const v4f q0 = __builtin_nontemporal_load(p + 0);
  const v4f q1 = __builtin_nontemporal_load(p + 1);
  const v4f q2 = __builtin_nontemporal_load(p + 2);
  const v4f q3 = __builtin_nontemporal_load(p + 3);
  const v4f q4 = __builtin_nontemporal_load(p + 4);

  const float scl[4] = { q0.x, q1.y, q2.z, q3.w };
  const float fs[4][4] = {
    { q0.y, q0.z, q0.w, q1.x },
    { q1.z, q1.w, q2.x, q2.y },
    { q2.w, q3.x, q3.y, q3.z },
    { q4.x, q4.y, q4.z, q4.w },
  };

  // v = sum_{s<len} scale_s * fs[s]
  float v[4] = { 0.f, 0.f, 0.f, 0.f };
#pragma unroll
  for (int s = 0; s < 4; ++s) {
    const float w = (s < len) ? scl[s] : 0.0f;
#pragma unroll
    for (int f = 0; f < 4; ++f) v[f] += w * fs[s][f];
  }

  // u = H v   (H uniform -> scalar loads)
  float u[4];
#pragma unroll
  for (int f = 0; f < 4; ++f)
    u[f] = H[f * 4 + 0] * v[0] + H[f * 4 + 1] * v[1] +
           H[f * 4 + 2] * v[2] + H[f * 4 + 3] * v[3];

  const int base = blockOff[blockIdx.x] + excl;
#pragma unroll
  for (int s = 0; s < 4; ++s) {
    if (s < len) {
      const float proj = fs[s][0] * u[0] + fs[s][1] * u[1] +
                         fs[s][2] * u[2] + fs[s][3] * u[3];
      __builtin_nontemporal_store(proj, out + base + s);
    }
  }
}

// ---------------------------------------------------------------------------
extern "C" void kernel_launch(void* const* d_in, const int* in_sizes, int n_in,
                              void* d_out, int out_size, void* d_ws, size_t ws_size,
                              hipStream_t stream) {
  const float* xss   = (const float*)d_in[0];
  const int*   seq   = (const int*)d_in[1];   // int32 on device (JAX x64 off)
  const float* W_cob = (const float*)d_in[2];
  const float* W_reg = (const float*)d_in[3];
  float* out = (float*)d_out;

  const int Btot = in_sizes[1];
  const int nBlocks = (Btot + 255) / 256;

  float* H    = (float*)d_ws;                      // 16 floats
  int*   sums = (int*)((char*)d_ws + 64);          // nBlocks ints

  fold_weights_wmma<<<1, 32, 0, stream>>>(W_cob, W_reg, H);
  block_count<<<nBlocks, 256, 0, stream>>>(seq, sums, Btot);
  scan_blocks<<<1, 1024, 0, stream>>>(sums, nBlocks);
  particle_main<<<nBlocks, 256, 0, stream>>>(xss, seq, H, sums, out, Btot);
}